// PurePyTorchSKA_44899588113058
// MI455X (gfx1250) — compile-verified
//
#include <hip/hip_runtime.h>
#include <stdint.h>

// Problem sizes fixed by the reference.
#define BATCH   8
#define CTOT    256
#define WC      32
#define NGROUP  8
#define HH      64
#define WW      64
#define HW      (HH * WW)
#define TH      16                 // output rows per block tile
#define NTILES  (HH / TH)          // 4
#define LDS_R   (TH + 2)           // 18 staged rows (vertical halo)
#define LDS_STR 72                 // dword stride per staged row (16B multiple)
#define BUF_DW  (LDS_R * LDS_STR)  // 1296 dwords per buffer
#define NTASK   (LDS_R * 16)       // 288 aligned 16B interior chunks per tile

// CDNA5 async global->LDS copy (ASYNCcnt). vdst = per-lane LDS byte address,
// vaddr = 64-bit global address, GV mode (saddr = off).
__device__ __forceinline__ void async_ld_b128(uint32_t lds_off, const float* g) {
  asm volatile("global_load_async_to_lds_b128 %0, %1, off"
               :: "v"(lds_off), "v"(g) : "memory");
}

__device__ __forceinline__ void wait_async0() {
#if __has_builtin(__builtin_amdgcn_s_wait_asynccnt)
  __builtin_amdgcn_s_wait_asynccnt(0);
#else
  asm volatile("s_wait_asynccnt 0" ::: "memory");
#endif
}

__global__ __launch_bounds__(256, 2)
void ska_fwd(const float* __restrict__ x,
             const float* __restrict__ wt,
             float* __restrict__ out)
{
  // Two ping-pong staging buffers so the async DMA for group g+1 overlaps
  // the compute of group g.
  __shared__ __align__(16) float xt[2 * BUF_DW];

  const int tid  = threadIdx.x;
  const int bid  = blockIdx.x;
  const int tile = bid & (NTILES - 1);
  const int bc   = bid >> 2;        // b*32 + cc
  const int b    = bc >> 5;
  const int cc   = bc & 31;
  const int h0   = tile * TH;

  const int tw = tid & 15;          // w-quad (16 quads cover W=64)
  const int th = tid >> 4;          // row within tile (0..15)
  const int w0 = tw << 2;
  const int h  = h0 + th;

  // 9 per-pixel taps for this thread's 4 output columns, held in VGPRs
  // (9 x float4 = 36 VGPRs) and reused by all 8 groups: weights are read
  // from HBM exactly once (the reference reads them 8x).
  float4 wr[9];
  {
    const float* wb = wt + ((size_t)(b * WC + cc) * 9) * HW + h * WW + w0;
#pragma unroll
    for (int t = 0; t < 9; ++t)
      wr[t] = *reinterpret_cast<const float4*>(wb + (size_t)t * HW);
  }

  // Zero both buffers once: this permanently provides the padding halo
  // (left/right columns, out-of-image rows). The per-group DMA only ever
  // overwrites in-bounds interior chunks, so zeros never need refreshing.
  for (int i = tid; i < 2 * BUF_DW; i += 256) xt[i] = 0.0f;

  // Flat pointer into the LDS aperture: low 32 bits == LDS byte offset.
  const uint32_t lds_base = (uint32_t)(uintptr_t)(&xt[0]);

  // Precompute this thread's <=2 staging tasks (constant across groups).
  // Task space: 288 = 18 rows x 16 aligned 16B chunks; interior x col gw
  // maps to LDS col gw+4 so every chunk is 16B aligned.
  int      goff0 = -1, goff1 = -1;     // element offset within an x plane
  uint32_t loff0 = 0,  loff1 = 0;      // byte offset within a staging buffer
  {
    const int r0 = tid >> 4, q0 = tid & 15;
    const int gh0 = h0 - 1 + r0;
    if ((unsigned)gh0 < HH) {
      goff0 = gh0 * WW + (q0 << 2);
      loff0 = (uint32_t)(r0 * LDS_STR + 4 + (q0 << 2)) * 4u;
    }
    if (tid < NTASK - 256) {           // tasks 256..287
      const int s1 = tid + 256;
      const int r1 = s1 >> 4, q1 = s1 & 15;
      const int gh1 = h0 - 1 + r1;
      if ((unsigned)gh1 < HH) {
        goff1 = gh1 * WW + (q1 << 2);
        loff1 = (uint32_t)(r1 * LDS_STR + 4 + (q1 << 2)) * 4u;
      }
    }
  }

  const float* xbase = x + (size_t)(b * CTOT + cc) * HW;  // + g*WC*HW per group

  // Order the zero-fill DS stores before the first DMA, then prime buffer 0.
  __syncthreads();
  if (goff0 >= 0) async_ld_b128(lds_base + loff0, xbase + goff0);
  if (goff1 >= 0) async_ld_b128(lds_base + loff1, xbase + goff1);
  wait_async0();
  __syncthreads();

  for (int g = 0; g < NGROUP; ++g) {
    // Kick off the DMA for the next group into the other buffer; it runs
    // concurrently with this group's compute below.
    if (g + 1 < NGROUP) {
      const float*   xn  = xbase + (size_t)(g + 1) * WC * HW;
      const uint32_t nxt = (uint32_t)((g + 1) & 1) * (BUF_DW * 4u);
      if (goff0 >= 0) async_ld_b128(lds_base + nxt + loff0, xn + goff0);
      if (goff1 >= 0) async_ld_b128(lds_base + nxt + loff1, xn + goff1);
    }

    // Each x element is reused 9x out of LDS.
    const float* buf = xt + (g & 1) * BUF_DW;
    float a0 = 0.f, a1 = 0.f, a2 = 0.f, a3 = 0.f;
#pragma unroll
    for (int ki = 0; ki < 3; ++ki) {
      const float* row = &buf[(th + ki) * LDS_STR + w0 + 3];
      const float r0 = row[0], r1 = row[1], r2 = row[2],
                  r3 = row[3], r4 = row[4], r5 = row[5];
      const float4 wa = wr[3 * ki + 0];
      const float4 wb = wr[3 * ki + 1];
      const float4 wc = wr[3 * ki + 2];
      a0 = fmaf(r0, wa.x, fmaf(r1, wb.x, fmaf(r2, wc.x, a0)));
      a1 = fmaf(r1, wa.y, fmaf(r2, wb.y, fmaf(r3, wc.y, a1)));
      a2 = fmaf(r2, wa.z, fmaf(r3, wb.z, fmaf(r4, wc.z, a2)));
      a3 = fmaf(r3, wa.w, fmaf(r4, wb.w, fmaf(r5, wc.w, a3)));
    }

    float4 o = {a0, a1, a2, a3};
    *reinterpret_cast<float4*>(
        out + (size_t)(b * CTOT + g * WC + cc) * HW + h * WW + w0) = o;

    // Next buffer must be resident (each wave waits its own DMAs, barrier
    // makes it collective) and everyone must be done reading the current
    // buffer before iteration g+1 stages into it.
    wait_async0();
    __syncthreads();
  }
}

extern "C" void kernel_launch(void* const* d_in, const int* in_sizes, int n_in,
                              void* d_out, int out_size, void* d_ws, size_t ws_size,
                              hipStream_t stream) {
  const float* x  = (const float*)d_in[0];   // (8,256,64,64) fp32
  const float* wt = (const float*)d_in[1];   // (8,32,9,64,64) fp32
  float* out = (float*)d_out;                // (8,256,64,64) fp32
  (void)in_sizes; (void)n_in; (void)d_ws; (void)ws_size; (void)out_size;

  dim3 grid(BATCH * WC * NTILES);  // 1024 blocks
  dim3 block(256);                 // 8 wave32s
  hipLaunchKernelGGL(ska_fwd, grid, block, 0, stream, x, wt, out);
}